// GINConvNet_80539226735230
// MI455X (gfx1250) — compile-verified
//
#include <hip/hip_runtime.h>
#include <hip/hip_bf16.h>

typedef __attribute__((ext_vector_type(2))) float v2f;
typedef __attribute__((ext_vector_type(8))) float v8f;

#define N_NODES 520192
#define N_EDGES 4161536
#define N_GRAPHS 2048
#define MAXLEN 254
#define DIM 64
#define NFEAT 78

// ---------------------------------------------------------------------------
// Wave-level 16x64 GEMM tile using CDNA5 fp32 WMMA (V_WMMA_F32_16X16X4_F32).
// A: 16 x K (row-major, stride lda) in LDS.  Bt: 64 x K (W transposed, stride
// ldb) in LDS.  acc[nb] holds the 16x16 f32 D fragment for columns nb*16..+15.
// C/D layout: VGPR r -> row r (lanes 0-15) / row r+8 (lanes 16-31), col = lane%16.
// A frag:  lane l<16 -> {z[l][k0], z[l][k0+1]};  lane l>=16 -> {z[l-16][k0+2], ...+3}
// B frag mirrors A's K placement (lanes 16-31 carry K+2).
// ---------------------------------------------------------------------------
__device__ __forceinline__ void wave_gemm(const float* __restrict__ A, int lda,
                                          const float* __restrict__ Bt, int ldb,
                                          const float* __restrict__ bias,
                                          v8f* acc, int K, int lane) {
  const int half = lane >> 4;
  const int l = lane & 15;
#pragma unroll
  for (int nb = 0; nb < 4; ++nb) {
    float bv = bias[nb * 16 + l];
#pragma unroll
    for (int r = 0; r < 8; ++r) acc[nb][r] = bv;
  }
  for (int k0 = 0; k0 < K; k0 += 4) {
    const int kk = k0 + half * 2;
    v2f a;
    a.x = A[l * lda + kk];
    a.y = A[l * lda + kk + 1];
#pragma unroll
    for (int nb = 0; nb < 4; ++nb) {
      const int col = nb * 16 + l;
      v2f b;
      b.x = Bt[col * ldb + kk];
      b.y = Bt[col * ldb + kk + 1];
      acc[nb] = __builtin_amdgcn_wmma_f32_16x16x4_f32(
          false, a, false, b, (short)0, acc[nb], false, false);
    }
  }
}

// ---------------------------------------------------------------------------
// Edge aggregation (layer 0, F=78): one thread per (edge, 2 features).
// Rows are 312 B (8 B aligned) -> aligned float2 gather + 2 f32 atomics.
// ---------------------------------------------------------------------------
__global__ __launch_bounds__(256) void agg_kernel78(
    const float* __restrict__ h, const int* __restrict__ src,
    const int* __restrict__ dst, float* __restrict__ agg, long long total) {
  long long gid = (long long)blockIdx.x * 256 + threadIdx.x;
  if (gid >= total) return;
  int e = (int)(gid / 39);
  int f = (int)(gid % 39) * 2;
  int s = src[e];
  int d = dst[e];
  float2 v = *(const float2*)&h[(size_t)s * NFEAT + f];
  float* a = &agg[(size_t)d * NFEAT + f];
  atomicAdd(a + 0, v.x);
  atomicAdd(a + 1, v.y);
}

// ---------------------------------------------------------------------------
// Edge aggregation (F=64): one thread per (edge, 4 features).
// global_load_b128 gather + 4 f32 atomics; src/dst fetched 4x less often.
// ---------------------------------------------------------------------------
__global__ __launch_bounds__(256) void agg_kernel64(
    const float* __restrict__ h, const int* __restrict__ src,
    const int* __restrict__ dst, float* __restrict__ agg, long long total) {
  long long gid = (long long)blockIdx.x * 256 + threadIdx.x;
  if (gid >= total) return;
  int e = (int)(gid >> 4);
  int f = (int)(gid & 15) << 2;
  int s = src[e];
  int d = dst[e];
  float4 v = *(const float4*)&h[(size_t)s * DIM + f];
  float* a = &agg[(size_t)d * DIM + f];
  atomicAdd(a + 0, v.x);
  atomicAdd(a + 1, v.y);
  atomicAdd(a + 2, v.z);
  atomicAdd(a + 3, v.w);
}

// ---------------------------------------------------------------------------
// Fused GIN MLP: z = h + agg; z1 = relu(z @ W1 + b1); z2 = relu(z1 @ W2 + b2)
// Also accumulates per-feature sum / sum^2 for BatchNorm.
// Block = 256 threads = 8 waves; each wave computes one 16-row tile.
// ---------------------------------------------------------------------------
template <int F, int FPAD>
__global__ __launch_bounds__(256) void mlp_kernel(
    const float* __restrict__ h, const float* __restrict__ agg,
    const float* __restrict__ W1, const float* __restrict__ b1,
    const float* __restrict__ W2, const float* __restrict__ b2,
    float* __restrict__ z2out, float* __restrict__ colsum,
    float* __restrict__ colsumsq) {
  __shared__ float sWt1[64][FPAD];   // W1 transposed (col-major of W1)
  __shared__ float sWt2[64][64];     // W2 transposed
  __shared__ float sB1[64], sB2[64];
  __shared__ float sZ[8][16][FPAD];  // per-wave input tile
  __shared__ float sZ1[8][16][64];   // per-wave hidden tile
  __shared__ float sCsum[64], sCsumsq[64];

  const int tid = threadIdx.x;
  const int w = tid >> 5;
  const int lane = tid & 31;
  const int half = lane >> 4;
  const int l = lane & 15;

  // stage transposed weights (+ zero K padding) and biases
  for (int i = tid; i < 64 * FPAD; i += 256) {
    int c = i & 63, k = i >> 6;
    sWt1[c][k] = (k < F) ? W1[k * 64 + c] : 0.0f;
  }
  for (int i = tid; i < 64 * 64; i += 256) {
    int c = i & 63, k = i >> 6;
    sWt2[c][k] = W2[k * 64 + c];
  }
  if (tid < 64) {
    sB1[tid] = b1[tid];
    sB2[tid] = b2[tid];
    sCsum[tid] = 0.0f;
    sCsumsq[tid] = 0.0f;
  }

  // stage this wave's 16 x FPAD input tile: z = h + agg (pad cols -> 0)
  const int row0 = (blockIdx.x * 8 + w) * 16;
  for (int i = lane; i < 16 * FPAD; i += 32) {
    int r = i / FPAD, c = i % FPAD;
    float v = 0.0f;
    if (c < F) {
      size_t idx = (size_t)(row0 + r) * F + c;
      v = h[idx] + agg[idx];
    }
    sZ[w][r][c] = v;
  }
  __syncthreads();

  // GEMM1: [16 x FPAD] @ [FPAD x 64] (+b1), ReLU -> sZ1
  v8f acc[4];
  wave_gemm(&sZ[w][0][0], FPAD, &sWt1[0][0], FPAD, sB1, acc, FPAD, lane);
#pragma unroll
  for (int nb = 0; nb < 4; ++nb)
#pragma unroll
    for (int r = 0; r < 8; ++r)
      sZ1[w][r + half * 8][nb * 16 + l] = fmaxf(acc[nb][r], 0.0f);
  __syncthreads();

  // GEMM2: [16 x 64] @ [64 x 64] (+b2), ReLU, store + BN stats
  wave_gemm(&sZ1[w][0][0], 64, &sWt2[0][0], 64, sB2, acc, 64, lane);
#pragma unroll
  for (int nb = 0; nb < 4; ++nb) {
    const int col = nb * 16 + l;
    float s = 0.0f, sq = 0.0f;
#pragma unroll
    for (int r = 0; r < 8; ++r) {
      float v = fmaxf(acc[nb][r], 0.0f);
      s += v;
      sq += v * v;
      z2out[(size_t)(row0 + r + half * 8) * 64 + col] = v;
    }
    atomicAdd(&sCsum[col], s);
    atomicAdd(&sCsumsq[col], sq);
  }
  __syncthreads();
  if (tid < 64) {
    atomicAdd(&colsum[tid], sCsum[tid]);
    atomicAdd(&colsumsq[tid], sCsumsq[tid]);
  }
}

// ---------------------------------------------------------------------------
// BatchNorm finalize: per-feature scale/shift from batch stats
// ---------------------------------------------------------------------------
__global__ void bn_finalize(const float* __restrict__ colsum,
                            const float* __restrict__ colsumsq,
                            const float* __restrict__ gamma,
                            const float* __restrict__ beta,
                            float* __restrict__ scale,
                            float* __restrict__ shift, float invN) {
  int c = threadIdx.x;
  float mean = colsum[c] * invN;
  float var = colsumsq[c] * invN - mean * mean;
  float sc = gamma[c] * rsqrtf(var + 1e-5f);
  scale[c] = sc;
  shift[c] = beta[c] - mean * sc;
}

// ---------------------------------------------------------------------------
// Apply BN affine: out = z * scale[c] + shift[c]   (float4 vectorized)
// ---------------------------------------------------------------------------
__global__ __launch_bounds__(256) void bn_apply(const float4* __restrict__ z,
                                                const float* __restrict__ scale,
                                                const float* __restrict__ shift,
                                                float4* __restrict__ out,
                                                int total4) {
  int i = blockIdx.x * 256 + threadIdx.x;
  if (i >= total4) return;
  int c = (i * 4) & 63;
  float4 v = z[i];
  float4 o;
  o.x = v.x * scale[c + 0] + shift[c + 0];
  o.y = v.y * scale[c + 1] + shift[c + 1];
  o.z = v.z * scale[c + 2] + shift[c + 2];
  o.w = v.w * scale[c + 3] + shift[c + 3];
  out[i] = o;
}

// ---------------------------------------------------------------------------
// Per-graph sum pool over 254 contiguous rows (block per graph)
// ---------------------------------------------------------------------------
__global__ __launch_bounds__(256) void pool_kernel(const float* __restrict__ h,
                                                   float* __restrict__ pooled) {
  __shared__ float part[4][64];
  int g = blockIdx.x;
  int c = threadIdx.x & 63;
  int rg = threadIdx.x >> 6;
  float s = 0.0f;
  for (int r = rg; r < MAXLEN; r += 4)
    s += h[((size_t)g * MAXLEN + r) * 64 + c];
  part[rg][c] = s;
  __syncthreads();
  if (threadIdx.x < 64)
    pooled[(size_t)g * 64 + threadIdx.x] = part[0][threadIdx.x] +
                                           part[1][threadIdx.x] +
                                           part[2][threadIdx.x] +
                                           part[3][threadIdx.x];
}

// ---------------------------------------------------------------------------
// FC: y = relu(pooled @ W + b)  (WMMA, wave per 16-row tile)
// ---------------------------------------------------------------------------
__global__ __launch_bounds__(256) void fc_kernel(const float* __restrict__ pooled,
                                                 const float* __restrict__ W,
                                                 const float* __restrict__ bias,
                                                 float* __restrict__ ybuf) {
  __shared__ float sWt[64][64];
  __shared__ float sB[64];
  __shared__ float sA[8][16][64];
  const int tid = threadIdx.x;
  for (int i = tid; i < 64 * 64; i += 256) {
    int c = i & 63, k = i >> 6;
    sWt[c][k] = W[k * 64 + c];
  }
  if (tid < 64) sB[tid] = bias[tid];
  const int w = tid >> 5, lane = tid & 31;
  const int half = lane >> 4, l = lane & 15;
  const int row0 = (blockIdx.x * 8 + w) * 16;
  for (int i = lane; i < 16 * 64; i += 32) {
    int r = i >> 6, c = i & 63;
    sA[w][r][c] = pooled[(size_t)(row0 + r) * 64 + c];
  }
  __syncthreads();
  v8f acc[4];
  wave_gemm(&sA[w][0][0], 64, &sWt[0][0], 64, sB, acc, 64, lane);
#pragma unroll
  for (int nb = 0; nb < 4; ++nb) {
    const int col = nb * 16 + l;
#pragma unroll
    for (int r = 0; r < 8; ++r)
      ybuf[(size_t)(row0 + r + half * 8) * 64 + col] = fmaxf(acc[nb][r], 0.0f);
  }
}

// ---------------------------------------------------------------------------
// LayerNorm over dim=64, one wave per row (wave32: 2 cols/lane, shfl_xor tree)
// ---------------------------------------------------------------------------
__global__ __launch_bounds__(256) void ln_kernel(const float* __restrict__ y,
                                                 const float* __restrict__ gamma,
                                                 const float* __restrict__ beta,
                                                 float* __restrict__ out) {
  const int w = threadIdx.x >> 5, lane = threadIdx.x & 31;
  const int row = blockIdx.x * 8 + w;
  float v0 = y[(size_t)row * 64 + lane];
  float v1 = y[(size_t)row * 64 + 32 + lane];
  float s = v0 + v1;
  float sq = v0 * v0 + v1 * v1;
#pragma unroll
  for (int m = 16; m >= 1; m >>= 1) {
    s += __shfl_xor(s, m, 32);
    sq += __shfl_xor(sq, m, 32);
  }
  float mu = s * (1.0f / 64.0f);
  float var = sq * (1.0f / 64.0f) - mu * mu;
  float inv = rsqrtf(var + 1e-5f);
  out[(size_t)row * 64 + lane] = (v0 - mu) * inv * gamma[lane] + beta[lane];
  out[(size_t)row * 64 + 32 + lane] =
      (v1 - mu) * inv * gamma[32 + lane] + beta[32 + lane];
}

// ---------------------------------------------------------------------------
// Host launcher
// ---------------------------------------------------------------------------
extern "C" void kernel_launch(void* const* d_in, const int* in_sizes, int n_in,
                              void* d_out, int out_size, void* d_ws,
                              size_t ws_size, hipStream_t stream) {
  const float* x = (const float*)d_in[0];
  const int* ei = (const int*)d_in[1];
  const float* w1a = (const float*)d_in[5];
  const float* b1a = (const float*)d_in[6];
  const float* w1b = (const float*)d_in[7];
  const float* b1b = (const float*)d_in[8];
  const float* Wa = (const float*)d_in[9];
  const float* ba = (const float*)d_in[10];
  const float* Wb = (const float*)d_in[11];
  const float* bb = (const float*)d_in[12];
  const float* bn_gamma = (const float*)d_in[13];
  const float* bn_beta = (const float*)d_in[14];
  const float* fc_w = (const float*)d_in[15];
  const float* fc_b = (const float*)d_in[16];
  const float* ln_gamma = (const float*)d_in[17];
  const float* ln_beta = (const float*)d_in[18];

  const int* src = ei;
  const int* dst = ei + N_EDGES;

  // workspace carve
  float* ws = (float*)d_ws;
  float* agg = ws;                              // N * 80
  float* zbuf = agg + (size_t)N_NODES * 80;     // N * 64
  float* hbuf = zbuf + (size_t)N_NODES * 64;    // N * 64
  float* stats = hbuf + (size_t)N_NODES * 64;   // 256
  float* colsum = stats;
  float* colsumsq = stats + 64;
  float* bnscale = stats + 128;
  float* bnshift = stats + 192;
  float* pooled = stats + 256;                  // B * 64
  float* ybuf = pooled + (size_t)N_GRAPHS * 64; // B * 64

  float* xseq_out = (float*)d_out;              // N * 64
  float* y_out = xseq_out + (size_t)N_NODES * 64;

  const float invN = 1.0f / (float)N_NODES;
  const int mlpBlocks = N_NODES / 16 / 8;        // 4064
  const int bnBlocks = (N_NODES * 64 / 4) / 256; // 32512

  const float* hcur = x;
  for (int layer = 0; layer < 5; ++layer) {
    const int F = (layer == 0) ? NFEAT : DIM;
    const float* W1 = (layer == 0) ? w1a : Wa + (size_t)(layer - 1) * 64 * 64;
    const float* B1 = (layer == 0) ? b1a : ba + (size_t)(layer - 1) * 64;
    const float* W2 = (layer == 0) ? w1b : Wb + (size_t)(layer - 1) * 64 * 64;
    const float* B2 = (layer == 0) ? b1b : bb + (size_t)(layer - 1) * 64;

    hipMemsetAsync(agg, 0, (size_t)N_NODES * F * sizeof(float), stream);
    if (layer == 0) {
      long long total = (long long)N_EDGES * 39;  // (edge, 2-feature) pairs
      int blocks = (int)((total + 255) / 256);
      agg_kernel78<<<blocks, 256, 0, stream>>>(hcur, src, dst, agg, total);
    } else {
      long long total = (long long)N_EDGES * 16;  // (edge, 4-feature) pairs
      int blocks = (int)((total + 255) / 256);
      agg_kernel64<<<blocks, 256, 0, stream>>>(hcur, src, dst, agg, total);
    }

    hipMemsetAsync(stats, 0, 128 * sizeof(float), stream);
    if (layer == 0)
      mlp_kernel<NFEAT, 80><<<mlpBlocks, 256, 0, stream>>>(
          hcur, agg, W1, B1, W2, B2, zbuf, colsum, colsumsq);
    else
      mlp_kernel<DIM, DIM><<<mlpBlocks, 256, 0, stream>>>(
          hcur, agg, W1, B1, W2, B2, zbuf, colsum, colsumsq);

    bn_finalize<<<1, 64, 0, stream>>>(colsum, colsumsq, bn_gamma + layer * 64,
                                      bn_beta + layer * 64, bnscale, bnshift,
                                      invN);
    float* hout = (layer == 4) ? xseq_out : hbuf;
    bn_apply<<<bnBlocks, 256, 0, stream>>>((const float4*)zbuf, bnscale,
                                           bnshift, (float4*)hout,
                                           N_NODES * 64 / 4);
    hcur = hout;
  }

  // pooled = per-graph sum over xseq_out; y = layernorm(relu(pooled@fc + b))
  pool_kernel<<<N_GRAPHS, 256, 0, stream>>>(xseq_out, pooled);
  fc_kernel<<<N_GRAPHS / 128, 256, 0, stream>>>(pooled, fc_w, fc_b, ybuf);
  ln_kernel<<<N_GRAPHS / 8, 256, 0, stream>>>(ybuf, ln_gamma, ln_beta, y_out);
}